// YOLOv1Loss_5299989643876
// MI455X (gfx1250) — compile-verified
//
#include <hip/hip_runtime.h>

typedef __attribute__((ext_vector_type(2))) float v2f;
typedef __attribute__((ext_vector_type(4))) float v4f;
typedef __attribute__((ext_vector_type(8))) float v8f;

#define SS        14
#define NBATCH    4096
#define CELLS     (NBATCH * SS * SS)     // 802816
#define TPB       256
#define NWG       (CELLS / TPB)          // 3136 = 49 * 64
#define CH        30
#define CHUNK_F   (TPB * CH)             // 7680 floats per tensor per block
#define CHUNK_V4  (CHUNK_F / 4)          // 1920 float4

__device__ __forceinline__ float box_iou(float px, float py, float pw, float ph,
                                         float tx, float ty, float tw, float th) {
    const float inv14 = 1.0f / 14.0f;
    float px1 = px * inv14 - 0.5f * pw, px2 = px * inv14 + 0.5f * pw;
    float py1 = py * inv14 - 0.5f * ph, py2 = py * inv14 + 0.5f * ph;
    float tx1 = tx * inv14 - 0.5f * tw, tx2 = tx * inv14 + 0.5f * tw;
    float ty1 = ty * inv14 - 0.5f * th, ty2 = ty * inv14 + 0.5f * th;
    float ix = fmaxf(fminf(px2, tx2) - fmaxf(px1, tx1), 0.0f);
    float iy = fmaxf(fminf(py2, ty2) - fmaxf(py1, ty1), 0.0f);
    float inter = ix * iy;
    float uni = pw * ph + tw * th - inter;
    return inter / uni;
}

__global__ __launch_bounds__(TPB)
void yolo_loss_main(const float* __restrict__ pred,
                    const float* __restrict__ tgt,
                    float* __restrict__ partial) {
    __shared__ float sm[2 * CHUNK_F];            // 61440 B
    const int tid = threadIdx.x;
    const int wg  = blockIdx.x;

    // ---- Stage 256 cells of pred+tgt into LDS: coalesced, non-temporal b128 ----
    const v4f* gp = (const v4f*)(pred + (size_t)wg * CHUNK_F);
    const v4f* gt = (const v4f*)(tgt  + (size_t)wg * CHUNK_F);
    v4f* s4 = (v4f*)sm;
    for (int i = tid; i < CHUNK_V4; i += TPB) s4[i]            = __builtin_nontemporal_load(&gp[i]);
    for (int i = tid; i < CHUNK_V4; i += TPB) s4[CHUNK_V4 + i] = __builtin_nontemporal_load(&gt[i]);
    __syncthreads();

    // ---- Per-cell loss ----
    const float* P = sm + tid * CH;
    const float* T = sm + CHUNK_F + tid * CH;
    float p[10], t[10];
#pragma unroll
    for (int i = 0; i < 10; ++i) { p[i] = P[i]; t[i] = T[i]; }

    float coo = (t[4] > 0.0f) ? 1.0f : 0.0f;
    float noo = 1.0f - coo;

    float d4 = p[4] - t[4], d9 = p[9] - t[9];
    float noo_l = noo * (d4 * d4 + d9 * d9);

    float cls = 0.0f;
#pragma unroll
    for (int c = 10; c < 30; ++c) { float d = P[c] - T[c]; cls = fmaf(d, d, cls); }
    cls *= coo;

    // IoU of both pred boxes vs. target box 0
    float iou0 = box_iou(p[0], p[1], p[2], p[3], t[0], t[1], t[2], t[3]);
    float iou1 = box_iou(p[5], p[6], p[7], p[8], t[0], t[1], t[2], t[3]);
    bool  b1   = iou1 > iou0;                    // argmax tie -> box 0 (matches jnp.argmax)
    float miou = fmaxf(iou0, iou1);

    float rx = b1 ? p[5] : p[0], ry = b1 ? p[6] : p[1];
    float rw = b1 ? p[7] : p[2], rh = b1 ? p[8] : p[3];
    float rc = b1 ? p[9] : p[4];
    float nc = b1 ? p[4] : p[9];                 // non-responsible confidence
    float tx = b1 ? t[5] : t[0], ty = b1 ? t[6] : t[1];
    float tw = b1 ? t[7] : t[2], th = b1 ? t[8] : t[3];

    float dx = rx - tx, dy = ry - ty;
    float dw = sqrtf(rw) - sqrtf(tw);
    float dh = sqrtf(rh) - sqrtf(th);
    float loc = coo * (dx * dx + dy * dy + dw * dw + dh * dh);

    float dc      = rc - miou;
    float contain = coo * dc * dc;
    float notc    = coo * nc * nc;
    float rest    = 2.0f * contain + notc + 0.5f * noo_l;

    // ---- Deterministic block reduction of (loc, cls, rest) ----
    __syncthreads();                             // done reading cell data from LDS
    float* rA = sm; float* rB = sm + TPB; float* rC = sm + 2 * TPB;
    rA[tid] = loc; rB[tid] = cls; rC[tid] = rest;
    __syncthreads();
    for (int s = TPB / 2; s > 0; s >>= 1) {
        if (tid < s) {
            rA[tid] += rA[tid + s];
            rB[tid] += rB[tid + s];
            rC[tid] += rC[tid + s];
        }
        __syncthreads();
    }
    if (tid == 0) {
        partial[0 * NWG + wg] = rA[0];
        partial[1 * NWG + wg] = rB[0];
        partial[2 * NWG + wg] = rC[0];
    }
}

// Final deterministic reduction of 3 x 3136 partials using V_WMMA_F32_16X16X4_F32.
// Each 64-value chunk -> A (16x4 f32, v2f/lane); B = ones(4x16); D accumulates
// row sums over 49 chunks. Grand total = lane0 regsum + lane16 regsum.
__global__ __launch_bounds__(32)
void yolo_loss_final(const float* __restrict__ partial, float* __restrict__ out) {
    const int lane = threadIdx.x;
    v2f ones; ones.x = 1.0f; ones.y = 1.0f;
    float totals[3];
#pragma unroll
    for (int c = 0; c < 3; ++c) {
        const float* base = partial + c * NWG;
        v8f acc = {};
        for (int k = 0; k < NWG / 64; ++k) {     // 49 chunks of 64
            v2f a;
            a.x = base[k * 64 + lane];
            a.y = base[k * 64 + 32 + lane];
            acc = __builtin_amdgcn_wmma_f32_16x16x4_f32(
                false, a, false, ones, (short)0, acc, false, false);
        }
        float s = acc[0] + acc[1] + acc[2] + acc[3] + acc[4] + acc[5] + acc[6] + acc[7];
        totals[c] = __shfl(s, 0, 32) + __shfl(s, 16, 32);
    }
    if (lane == 0) {
        float loc5 = 5.0f * totals[0];           // L_COORD * loc_loss
        float cls  = totals[1];
        float rest = totals[2];                  // 2*contain + not_contain + 0.5*noobj
        out[0] = (loc5 + rest + cls) / (float)NBATCH;
        out[1] = loc5;
        out[2] = cls;
    }
}

extern "C" void kernel_launch(void* const* d_in, const int* in_sizes, int n_in,
                              void* d_out, int out_size, void* d_ws, size_t ws_size,
                              hipStream_t stream) {
    const float* pred = (const float*)d_in[0];
    const float* tgt  = (const float*)d_in[1];
    float* partial = (float*)d_ws;               // needs 3*3136*4 = 37632 B
    float* out = (float*)d_out;

    yolo_loss_main<<<NWG, TPB, 0, stream>>>(pred, tgt, partial);
    yolo_loss_final<<<1, 32, 0, stream>>>(partial, out);
}